// Model_48936857370759
// MI455X (gfx1250) — compile-verified
//
#include <hip/hip_runtime.h>
#include <stdint.h>

// ---------------- problem constants (from reference) ----------------
#define N_E   200000
#define N_R   1000000
#define N_Q   50000
#define E_P   1000000
#define E_U   1000000
#define D_L   64
#define D_H   128
#define DIM   192          // D_L + D_H
#define C48   48           // float4 chunks per 192-f row
#define C32   32           // float4 chunks per 128-f row
#define C16   16           // float4 chunks per 64-f row

// edge-kernel tiling
#define TILE  256          // edges staged per block (both edge kernels)
#define EPR   8            // edges per round
#define BLK5  384          // k_conv: 8 edges x 48 chunks
#define BLK3  256          // k_prof: 8 edges x 32 chunks

typedef float        f4  __attribute__((ext_vector_type(4)));
typedef unsigned int v4u __attribute__((ext_vector_type(4)));
typedef int          v4i __attribute__((ext_vector_type(4)));
typedef int          v8i __attribute__((ext_vector_type(8)));

// Hardware f32 atomic add -> global_atomic_add_f32 (no CAS loop).
__device__ __forceinline__ void atom_add_f32(float* p, float v) {
    unsafeAtomicAdd(p, v);
}

// ---------------- TDM: async 1-D global->LDS copy of 4B elements ----------------
// Packs a D# per CDNA5 ISA (group0: count=1, lds_addr,
// global_addr, type=2; group1: data_size=4B, tensor_dim0, tile_dim0, dim1=1).
// OOB-of-tensor reads return zero, additionally guarded in the consumer.
#if __has_builtin(__builtin_amdgcn_tensor_load_to_lds)
#define HAVE_TDM 1
__device__ __forceinline__ void tdm_load_1d(unsigned int lds_addr, const void* gptr,
                                            unsigned int tensor_elems,
                                            unsigned int tile_elems) {
    unsigned long long ga = (unsigned long long)(uintptr_t)gptr;
    v4u g0 = {0u, 0u, 0u, 0u};
    g0[0] = 1u;                                   // count=1 (valid user descriptor)
    g0[1] = lds_addr;                             // LDS byte address
    g0[2] = (unsigned int)ga;                     // global_addr[31:0]
    g0[3] = (unsigned int)((ga >> 32) & 0x01FFFFFFull) | 0x80000000u; // [56:32] | type=2
    v8i g1 = {0, 0, 0, 0, 0, 0, 0, 0};
    g1[0] = (int)(2u << 16);                                      // data_size = 4B
    g1[1] = (int)((tensor_elems & 0xFFFFu) << 16);                // tensor_dim0 lo16 @bit48
    g1[2] = (int)(((tensor_elems >> 16) & 0xFFFFu) | (1u << 16)); // dim0 hi16 | tensor_dim1=1
    g1[3] = (int)((tile_elems & 0xFFFFu) << 16);                  // tile_dim0 @bit112
    g1[4] = 1;                                                    // tile_dim1 = 1
    g1[5] = (int)tensor_elems;                                    // tensor_dim0_stride lo
    v4i z4 = {0, 0, 0, 0};
#if defined(__clang_major__) && (__clang_major__ >= 23)
    v8i z8 = {0, 0, 0, 0, 0, 0, 0, 0};
    __builtin_amdgcn_tensor_load_to_lds(g0, g1, z4, z4, z8, 0);
#else
    __builtin_amdgcn_tensor_load_to_lds(g0, g1, z4, z4, 0);
#endif
}
#else
#define HAVE_TDM 0
#endif

// ---------------- K5 (emitted first so disasm snippet shows the hot kernel) -------
// One conv step's scatter work. Block stages its TILE edge indices (3 arrays) into
// LDS via the Tensor Data Mover, waits on TENSORcnt, then 8 edges x 48 float4-chunk
// threads gather e/query rows (global_load_b128, L2-temporal) and scatter with
// hardware f32 atomics.
__global__ __launch_bounds__(BLK5) void k_conv(const float* __restrict__ e_cur,
                                               float* __restrict__ e_new,
                                               const float* __restrict__ query_e0,
                                               const float* __restrict__ inv,
                                               const int* __restrict__ pur_src,
                                               const int* __restrict__ pur_dst,
                                               const int* __restrict__ q_id) {
    __shared__ int sidx[3 * TILE];
    const int base = blockIdx.x * TILE;
#if HAVE_TDM
    if ((threadIdx.x >> 5) == 0) {        // wave 0 issues the three DMA descriptors
        unsigned int rem  = (unsigned int)(E_U - base);
        unsigned int lds0 = (unsigned int)(uintptr_t)&sidx[0]; // low 32b = LDS offset
        tdm_load_1d(lds0,                pur_src + base, rem, TILE);
        tdm_load_1d(lds0 + TILE * 4,     pur_dst + base, rem, TILE);
        tdm_load_1d(lds0 + 2 * TILE * 4, q_id   + base,  rem, TILE);
        __builtin_amdgcn_s_wait_tensorcnt(0);
    }
#else
    for (int t = threadIdx.x; t < TILE; t += BLK5) {
        int eg = base + t;
        sidx[t]            = (eg < E_U) ? pur_src[eg] : 0;
        sidx[TILE + t]     = (eg < E_U) ? pur_dst[eg] : 0;
        sidx[2 * TILE + t] = (eg < E_U) ? q_id[eg]    : 0;
    }
#endif
    __syncthreads();

    const int sub = (int)threadIdx.x / C48;   // 0..7  (edge within round)
    const int c   = (int)threadIdx.x % C48;   // 0..47 (float4 chunk of 192-f row)
    const f4* E4 = (const f4*)e_cur;
    const f4* Q4 = (const f4*)query_e0;

    for (int r = 0; r < TILE / EPR; ++r) {
        int el = r * EPR + sub;
        int eg = base + el;
        if (eg < E_U) {
            int s = sidx[el], d = sidx[TILE + el], q = sidx[2 * TILE + el];
            float is = inv[s], id = inv[d];
            // forward message: m = e[s] + query_e0[q] * inv[s]  ->  e_new[d]
            f4 es = E4[(size_t)s * C48 + c];
            f4 qv = Q4[(size_t)q * C48 + c];
            f4 m  = es + qv * is;
            float* pd = e_new + (size_t)d * DIM + c * 4;
            atom_add_f32(pd + 0, m.x); atom_add_f32(pd + 1, m.y);
            atom_add_f32(pd + 2, m.z); atom_add_f32(pd + 3, m.w);
            // reverse message: e_msg[d] = e[d] * inv[d]  ->  e_new[s]
            f4 ed = E4[(size_t)d * C48 + c] * id;
            float* ps = e_new + (size_t)s * DIM + c * 4;
            atom_add_f32(ps + 0, ed.x); atom_add_f32(ps + 1, ed.y);
            atom_add_f32(ps + 2, ed.z); atom_add_f32(ps + 3, ed.w);
        }
    }
}

// ---------------- K3: scatter review_h[src] into e[dst, 64:192] --------------------
// Same TDM-staged pattern: block stages TILE (src,dst) index pairs, then 8 edges x
// 32 float4-chunk threads stream review rows (NT loads: ~512 MB touched once, keep
// it out of the 192 MB L2 which we reserve for e/query tables) and scatter-add.
__global__ __launch_bounds__(BLK3) void k_prof(const float* __restrict__ review_h,
                                               const int* __restrict__ prof_src,
                                               const int* __restrict__ prof_dst,
                                               float* __restrict__ A) {
    __shared__ int pidx[2 * TILE];
    const int base = blockIdx.x * TILE;
#if HAVE_TDM
    if ((threadIdx.x >> 5) == 0) {
        unsigned int rem  = (unsigned int)(E_P - base);
        unsigned int lds0 = (unsigned int)(uintptr_t)&pidx[0];
        tdm_load_1d(lds0,            prof_src + base, rem, TILE);
        tdm_load_1d(lds0 + TILE * 4, prof_dst + base, rem, TILE);
        __builtin_amdgcn_s_wait_tensorcnt(0);
    }
#else
    for (int t = threadIdx.x; t < TILE; t += BLK3) {
        int eg = base + t;
        pidx[t]        = (eg < E_P) ? prof_src[eg] : 0;
        pidx[TILE + t] = (eg < E_P) ? prof_dst[eg] : 0;
    }
#endif
    __syncthreads();

    const int sub = (int)threadIdx.x / C32;   // 0..7
    const int c   = (int)threadIdx.x % C32;   // 0..31 (float4 chunk of 128-f row)

    for (int r = 0; r < TILE / EPR; ++r) {
        int el = r * EPR + sub;
        int eg = base + el;
        if (eg < E_P) {
            int s = pidx[el], d = pidx[TILE + el];
            f4 v = __builtin_nontemporal_load(((const f4*)review_h) + (size_t)s * C32 + c);
            float* dst = A + (size_t)d * DIM + D_L + c * 4;
            atom_add_f32(dst + 0, v.x); atom_add_f32(dst + 1, v.y);
            atom_add_f32(dst + 2, v.z); atom_add_f32(dst + 3, v.w);
        }
    }
}

// ---------------- K1: e[:, :64] = entity_l ; e[:, 64:] = 0 ; zero degrees ----------
__global__ void k_init(const float* __restrict__ entity_l, float* __restrict__ A,
                       float* __restrict__ deg_p, float* __restrict__ deg_i) {
    size_t gid = (size_t)blockIdx.x * blockDim.x + threadIdx.x;
    if (gid >= (size_t)N_E * C48) return;
    int i = (int)(gid / C48), c = (int)(gid % C48);
    f4 v = {0.f, 0.f, 0.f, 0.f};
    if (c < C16) v = ((const f4*)entity_l)[(size_t)i * C16 + c];
    ((f4*)A)[(size_t)i * C48 + c] = v;
    if (c == 0) { deg_p[i] = 0.f; deg_i[i] = 0.f; }
}

// ---------------- K2: degree histograms (streamed NT index reads) ------------------
__global__ void k_deg(const int* __restrict__ prof_dst, const int* __restrict__ pur_src,
                      const int* __restrict__ pur_dst, float* __restrict__ deg_p,
                      float* __restrict__ deg_i) {
    int t = blockIdx.x * blockDim.x + threadIdx.x;
    if (t < E_P) atom_add_f32(&deg_p[__builtin_nontemporal_load(&prof_dst[t])], 1.f);
    if (t < E_U) {
        atom_add_f32(&deg_i[__builtin_nontemporal_load(&pur_src[t])], 1.f);
        atom_add_f32(&deg_i[__builtin_nontemporal_load(&pur_dst[t])], 1.f);
    }
}

// ---------------- K4: normalize h by deg_p, inv = rsqrt(deg_i), out = e0, B = 0 ----
__global__ void k_norm(float* __restrict__ A, float* __restrict__ B, float* __restrict__ out,
                       const float* __restrict__ deg_p, float* __restrict__ inv_io) {
    size_t gid = (size_t)blockIdx.x * blockDim.x + threadIdx.x;
    if (gid >= (size_t)N_E * C48) return;
    int i = (int)(gid / C48), c = (int)(gid % C48);
    size_t idx = (size_t)i * C48 + c;
    f4 v = ((f4*)A)[idx];
    if (c >= C16) {                       // h-part: divide by clamped profile degree
        float r = 1.f / fmaxf(deg_p[i], 1.f);
        v = v * r;
        ((f4*)A)[idx] = v;
    }
    ((f4*)out)[idx] = v;                  // acc := e0
    f4 z = {0.f, 0.f, 0.f, 0.f};
    ((f4*)B)[idx] = z;                    // next e_new := 0
    if (c == 0) inv_io[i] = rsqrtf(fmaxf(inv_io[i], 1.f));
}

// ---------------- K6: e = e_new * inv ; acc += e ; zero the other buffer -----------
__global__ void k_upd(float* __restrict__ X, float* __restrict__ Y, float* __restrict__ out,
                      const float* __restrict__ inv, int final_) {
    size_t gid = (size_t)blockIdx.x * blockDim.x + threadIdx.x;
    if (gid >= (size_t)N_E * C48) return;
    int i = (int)(gid / C48), c = (int)(gid % C48);
    size_t idx = (size_t)i * C48 + c;
    f4 v = ((f4*)X)[idx] * inv[i];
    ((f4*)X)[idx] = v;                    // X now holds e_{k+1} for the next round
    f4 o = ((f4*)out)[idx] + v;
    if (final_) o = o * (1.f / 3.f);      // / (K_CONV + 1)
    ((f4*)out)[idx] = o;
    f4 z = {0.f, 0.f, 0.f, 0.f};
    ((f4*)Y)[idx] = z;                    // Y becomes the next e_new accumulator
}

// ---------------- launch -----------------------------------------------------------
extern "C" void kernel_launch(void* const* d_in, const int* in_sizes, int n_in,
                              void* d_out, int out_size, void* d_ws, size_t ws_size,
                              hipStream_t stream) {
    const float* entity_l = (const float*)d_in[0];
    const float* review_h = (const float*)d_in[1];
    const float* query_e0 = (const float*)d_in[2];
    const int*   prof_src = (const int*)d_in[3];
    const int*   prof_dst = (const int*)d_in[4];
    const int*   pur_src  = (const int*)d_in[5];
    const int*   pur_dst  = (const int*)d_in[6];
    const int*   q_id     = (const int*)d_in[7];
    float* out = (float*)d_out;

    // workspace: deg_p[N_E] | inv[N_E] | A[N_E*192] | B[N_E*192]   (~309 MB)
    float* deg_p = (float*)d_ws;
    float* invv  = deg_p + N_E;
    float* A     = invv + N_E;
    float* B     = A + (size_t)N_E * DIM;

    const int nrow_blk = (int)(((size_t)N_E * C48 + 255) / 256);
    const int nconvblk = (E_U + TILE - 1) / TILE;
    const int nprofblk = (E_P + TILE - 1) / TILE;

    k_init<<<nrow_blk, 256, 0, stream>>>(entity_l, A, deg_p, invv);
    k_deg <<<(E_U + 255) / 256, 256, 0, stream>>>(prof_dst, pur_src, pur_dst, deg_p, invv);
    k_prof<<<nprofblk, BLK3, 0, stream>>>(review_h, prof_src, prof_dst, A);
    k_norm<<<nrow_blk, 256, 0, stream>>>(A, B, out, deg_p, invv);

    // conv iteration 1: A -> B
    k_conv<<<nconvblk, BLK5, 0, stream>>>(A, B, query_e0, invv, pur_src, pur_dst, q_id);
    k_upd <<<nrow_blk, 256, 0, stream>>>(B, A, out, invv, 0);
    // conv iteration 2: B -> A
    k_conv<<<nconvblk, BLK5, 0, stream>>>(B, A, query_e0, invv, pur_src, pur_dst, q_id);
    k_upd <<<nrow_blk, 256, 0, stream>>>(A, B, out, invv, 1);
}